// YOLOFastestv2Head_86380382257159
// MI455X (gfx1250) — compile-verified
//
#include <hip/hip_runtime.h>

#define NUM_ANCHORS 3
#define NUM_CLASSES 80
#define C_IN 72

typedef __attribute__((ext_vector_type(2))) float v2f;
typedef __attribute__((ext_vector_type(8))) float v8f;

// LDS weight staging layout (rows x 72 floats, row-major):
//   rows 0..79   : w_cls
//   rows 80..95  : w_reg (12 valid rows, rest zero)
//   rows 96..111 : w_obj (3 valid rows, rest zero)
#define LDS_ROWS 112

template <int HW, int W>
__global__ __launch_bounds__(128) void yolo_head_kernel(
    const float* __restrict__ feat_reg,
    const float* __restrict__ feat_obj,
    const float* __restrict__ feat_cls,
    const float* __restrict__ w_reg, const float* __restrict__ b_reg,
    const float* __restrict__ w_obj, const float* __restrict__ b_obj,
    const float* __restrict__ w_cls, const float* __restrict__ b_cls,
    float* __restrict__ out,
    float stride_v, int level_off,
    float aw0, float ah0, float aw1, float ah1, float aw2, float ah2)
{
    __shared__ float smem[LDS_ROWS * C_IN];

    const int tid = threadIdx.x;

    // ---- Stage weights into LDS (coalesced), pad reg/obj rows with zeros ----
    for (int i = tid; i < NUM_CLASSES * C_IN; i += 128) smem[i] = w_cls[i];
    for (int i = tid; i < 16 * C_IN; i += 128) {
        int row = i / C_IN;
        smem[80 * C_IN + i] = (row < 12) ? w_reg[i] : 0.0f;
    }
    for (int i = tid; i < 16 * C_IN; i += 128) {
        int row = i / C_IN;
        smem[96 * C_IN + i] = (row < 3) ? w_obj[i] : 0.0f;
    }
    __syncthreads();

    const int wave = tid >> 5;
    const int lane = tid & 31;
    const int hi   = lane >> 4;   // which half of the wave
    const int n    = lane & 15;   // column index within 16-wide tile

    const int tile = blockIdx.x * 4 + wave;   // one 16-position tile per wave
    const int pos0 = tile * 16;               // HW % 16 == 0 -> tile within one image
    const int b    = pos0 / HW;               // constant divisor -> magic multiply
    const int hw0  = pos0 - b * HW;

    // Loop-invariant per-lane base pointers; per-k offsets are compile-time
    // constants (k*HW*4 bytes) that fold into the load's 24-bit IOFFSET.
    const size_t fbase = (size_t)b * C_IN * HW + hw0 + (size_t)(2 * hi) * HW + n;
    const float* frA = feat_reg + fbase;
    const float* foA = feat_obj + fbase;
    const float* fcA = feat_cls + fbase;

    // Loop-invariant per-lane LDS bases (v2f units); per-k offset = k0/2 v2f.
    const v2f* wsm = (const v2f*)smem;
    const v2f* wb0 = wsm + (((0 * 16 + n) * C_IN + 2 * hi) >> 1);
    const v2f* wb1 = wsm + (((1 * 16 + n) * C_IN + 2 * hi) >> 1);
    const v2f* wb2 = wsm + (((2 * 16 + n) * C_IN + 2 * hi) >> 1);
    const v2f* wb3 = wsm + (((3 * 16 + n) * C_IN + 2 * hi) >> 1);
    const v2f* wb4 = wsm + (((4 * 16 + n) * C_IN + 2 * hi) >> 1);
    const v2f* wbr = wsm + (((80 + n) * C_IN + 2 * hi) >> 1);
    const v2f* wbo = wsm + (((96 + n) * C_IN + 2 * hi) >> 1);

    v8f acc_c0 = {}, acc_c1 = {}, acc_c2 = {}, acc_c3 = {}, acc_c4 = {};
    v8f acc_r  = {}, acc_o  = {};

    // ---- K loop: 18 x V_WMMA_F32_16X16X4_F32 per head tile, fully unrolled;
    //      all load offsets are immediates, no per-iteration address math. ----
    #pragma unroll
    for (int k0 = 0; k0 < C_IN; k0 += 4) {
        v2f a_r, a_o, a_c;
        a_r.x = frA[(size_t)k0 * HW];  a_r.y = frA[(size_t)(k0 + 1) * HW];
        a_o.x = foA[(size_t)k0 * HW];  a_o.y = foA[(size_t)(k0 + 1) * HW];
        a_c.x = fcA[(size_t)k0 * HW];  a_c.y = fcA[(size_t)(k0 + 1) * HW];

        v2f b0 = wb0[k0 >> 1];
        v2f b1 = wb1[k0 >> 1];
        v2f b2 = wb2[k0 >> 1];
        v2f b3 = wb3[k0 >> 1];
        v2f b4 = wb4[k0 >> 1];
        v2f br = wbr[k0 >> 1];
        v2f bo = wbo[k0 >> 1];

        acc_c0 = __builtin_amdgcn_wmma_f32_16x16x4_f32(false, a_c, false, b0, (short)0, acc_c0, false, false);
        acc_c1 = __builtin_amdgcn_wmma_f32_16x16x4_f32(false, a_c, false, b1, (short)0, acc_c1, false, false);
        acc_c2 = __builtin_amdgcn_wmma_f32_16x16x4_f32(false, a_c, false, b2, (short)0, acc_c2, false, false);
        acc_c3 = __builtin_amdgcn_wmma_f32_16x16x4_f32(false, a_c, false, b3, (short)0, acc_c3, false, false);
        acc_c4 = __builtin_amdgcn_wmma_f32_16x16x4_f32(false, a_c, false, b4, (short)0, acc_c4, false, false);
        acc_r  = __builtin_amdgcn_wmma_f32_16x16x4_f32(false, a_r, false, br, (short)0, acc_r,  false, false);
        acc_o  = __builtin_amdgcn_wmma_f32_16x16x4_f32(false, a_o, false, bo, (short)0, acc_o,  false, false);
    }

    // ---- Epilogue: bias + sigmoid/softmax + decode + store ----
    float bc0 = b_cls[0 * 16 + n];
    float bc1 = b_cls[1 * 16 + n];
    float bc2 = b_cls[2 * 16 + n];
    float bc3 = b_cls[3 * 16 + n];
    float bc4 = b_cls[4 * 16 + n];
    float brv = (n < 12) ? b_reg[n] : 0.0f;
    float bov = (n < 3)  ? b_obj[n] : 0.0f;

    const float anw = (n < 4) ? aw0 : (n < 8) ? aw1 : aw2;  // anchor w for this lane's reg col
    const float anh = (n < 4) ? ah0 : (n < 8) ? ah1 : ah2;  // anchor h

    #pragma unroll
    for (int r = 0; r < 8; ++r) {
        // C-layout: VGPR r holds row M = r (lanes 0-15) / r+8 (lanes 16-31).
        const int hw_r = hw0 + r + 8 * hi;           // uniform per half-wave
        float* base = out + ((size_t)b * 6000 + (size_t)(level_off + hw_r * 3)) * 85;

        // --- class softmax across 80 logits (5 per lane x 16 lanes) ---
        float v0 = acc_c0[r] + bc0;
        float v1 = acc_c1[r] + bc1;
        float v2 = acc_c2[r] + bc2;
        float v3 = acc_c3[r] + bc3;
        float v4 = acc_c4[r] + bc4;
        float m = fmaxf(fmaxf(fmaxf(v0, v1), fmaxf(v2, v3)), v4);
        #pragma unroll
        for (int mask = 1; mask < 16; mask <<= 1)
            m = fmaxf(m, __shfl_xor(m, mask));
        float e0 = __expf(v0 - m), e1 = __expf(v1 - m), e2 = __expf(v2 - m);
        float e3 = __expf(v3 - m), e4 = __expf(v4 - m);
        float s = e0 + e1 + e2 + e3 + e4;
        #pragma unroll
        for (int mask = 1; mask < 16; mask <<= 1)
            s += __shfl_xor(s, mask);
        const float inv = 1.0f / s;
        const float p0 = e0 * inv, p1 = e1 * inv, p2 = e2 * inv, p3 = e3 * inv, p4 = e4 * inv;

        // class probs shared across the 3 anchors; half-wave-contiguous stores
        #pragma unroll
        for (int a = 0; a < NUM_ANCHORS; ++a) {
            float* cb = base + a * 85 + 5 + n;
            cb[0]  = p0;
            cb[16] = p1;
            cb[32] = p2;
            cb[48] = p3;
            cb[64] = p4;
        }

        // --- box regression: lanes 0..11 hold [anchor*4 + {x,y,w,h}] ---
        if (n < 12) {
            const int a    = n >> 2;
            const int comp = n & 3;
            const float sg = 1.0f / (1.0f + __expf(-(acc_r[r] + brv)));
            float val;
            if (comp < 2) {
                const float g = (comp == 0) ? (float)(hw_r % W) : (float)(hw_r / W);
                val = (sg * 2.0f - 0.5f + g) * stride_v;
            } else {
                const float t = sg * 2.0f;
                val = t * t * ((comp == 2) ? anw : anh);
            }
            base[a * 85 + comp] = val;
        }

        // --- objectness: lanes 0..2 ---
        if (n < 3) {
            const float so = 1.0f / (1.0f + __expf(-(acc_o[r] + bov)));
            base[n * 85 + 4] = so;
        }
    }
}

extern "C" void kernel_launch(void* const* d_in, const int* in_sizes, int n_in,
                              void* d_out, int out_size, void* d_ws, size_t ws_size,
                              hipStream_t stream) {
    (void)in_sizes; (void)n_in; (void)out_size; (void)d_ws; (void)ws_size;
    const float* fr0 = (const float*)d_in[0];
    const float* fo0 = (const float*)d_in[1];
    const float* fc0 = (const float*)d_in[2];
    const float* fr1 = (const float*)d_in[3];
    const float* fo1 = (const float*)d_in[4];
    const float* fc1 = (const float*)d_in[5];
    const float* wr  = (const float*)d_in[6];
    const float* br  = (const float*)d_in[7];
    const float* wo  = (const float*)d_in[8];
    const float* bo  = (const float*)d_in[9];
    const float* wc  = (const float*)d_in[10];
    const float* bc  = (const float*)d_in[11];
    float* out = (float*)d_out;

    // Level 0: 64*40*40 = 102400 positions -> 6400 tiles -> 1600 blocks (4 waves each)
    yolo_head_kernel<1600, 40><<<1600, 128, 0, stream>>>(
        fr0, fo0, fc0, wr, br, wo, bo, wc, bc, out,
        16.0f, /*level_off=*/0,
        12.64f, 19.39f, 37.88f, 51.48f, 55.71f, 138.31f);

    // Level 1: 64*20*20 = 25600 positions -> 1600 tiles -> 400 blocks
    yolo_head_kernel<400, 20><<<400, 128, 0, stream>>>(
        fr1, fo1, fc1, wr, br, wo, bo, wc, bc, out,
        32.0f, /*level_off=*/4800,
        126.91f, 78.23f, 131.57f, 214.55f, 279.92f, 258.87f);
}